// GraphLearningModule_14654428414231
// MI455X (gfx1250) — compile-verified
//
#include <hip/hip_runtime.h>
#include <math.h>

#define B_   2
#define T_   8
#define N_   5000
#define K1_  17
#define H_   4
#define C_   16
#define NO_  8
#define T7_  (T_-1)                 // 7
#define E2_  (N_*K1_)               // 85000 rows for part 2 per (b,t,h)
#define TILES2_ ((E2_+15)/16)       // 5313 (last tile has 8 padded rows)

typedef float v2f __attribute__((ext_vector_type(2)));
typedef float v8f __attribute__((ext_vector_type(8)));

// ds_swizzle_b32 xor-butterfly: lane gets value from lane^M (group-of-32 mode:
// offset[14:10]=xor_mask, offset[9:5]=or=0, offset[4:0]=and=0x1f)
#define SWZ(x, M) __int_as_float(__builtin_amdgcn_ds_swizzle( \
                      __float_as_int(x), ((M) << 10) | 0x1f))

// Tree merge: registers a (rows with bit==0) / b (rows with bit==1), columns in
// lanes. Each lane keeps the register matching its own column bit M and adds
// the complementary partial fetched from lane^M. After levels 1,2,4 + a final
// xor-8 fold, lane l holds the full 16-column sum of row (l&7) (+8 for upper
// lane half, whose D registers carry rows 8..15).
template<int M>
__device__ __forceinline__ float mergepair(float a, float b, int c) {
    const float keep = (c & M) ? b : a;
    const float send = (c & M) ? a : b;
    return keep + SWZ(send, M);
}

__device__ __forceinline__ float row_reduce16(const float t[8], int c) {
    const float m0 = mergepair<1>(t[0], t[1], c);
    const float m1 = mergepair<1>(t[2], t[3], c);
    const float m2 = mergepair<1>(t[4], t[5], c);
    const float m3 = mergepair<1>(t[6], t[7], c);
    const float n0 = mergepair<2>(m0, m1, c);
    const float n1 = mergepair<2>(m2, m3, c);
    float s = mergepair<4>(n0, n1, c);
    s += SWZ(s, 8);
    return s;   // lane l: sum over 16 cols of row (l&7)+8*(l>>4)
}

// ---------------------------------------------------------------------------
// Kernel 1: per (b*T+t, h, n) wave computes 16 neighbor rows:
//   df = f_i - f_j ;  Mdf = df @ multiM[h]^T via 4 chained V_WMMA_F32_16X16X4_F32
//   w  = exp(-||Mdf||^2) * valid ;  deg = sum_k w
// A layout (16x4 f32): lane = row | (K-half<<4), VGPR0/1 hold K = 2*hh + {0,1}
// B layout (4x16 f32): lane = col | (K-half<<4), B[j][i] = M[i][j]
// ---------------------------------------------------------------------------
__global__ void k1_w_deg(const float* __restrict__ feat,
                         const int*   __restrict__ nn,
                         const float* __restrict__ mm,
                         float*       __restrict__ w_out,
                         float*       __restrict__ deg)
{
    const int wid  = (blockIdx.x * blockDim.x + threadIdx.x) >> 5;  // grid exact
    const int lane = threadIdx.x & 31;

    const int n  = wid % N_;
    const int r0 = wid / N_;
    const int h  = r0 % H_;
    const int bt = r0 / H_;               // b*T + t, 0..15

    const int r  = lane & 15;             // row (neighbor slot) / B col
    const int hh = lane >> 4;             // K half

    const int nbrv = nn[n * K1_ + 1 + r];
    const unsigned vm = (unsigned)__ballot(nbrv >= 0);
    const int nj = (nbrv < 0) ? n : nbrv;  // invalid -> self (df=0, masked later)

    const float* fi = feat + ((((size_t)bt * N_ + n ) * H_ + h) * C_);
    const float* fj = feat + ((((size_t)bt * N_ + nj) * H_ + h) * C_);
    const float* Mh = mm + h * C_ * C_;

    v8f d = {0.f,0.f,0.f,0.f,0.f,0.f,0.f,0.f};
#pragma unroll
    for (int c = 0; c < 4; ++c) {
        const int j0 = 4 * c + 2 * hh;
        v2f av, bv;
        av.x = fi[j0]     - fj[j0];
        av.y = fi[j0 + 1] - fj[j0 + 1];
        bv.x = Mh[r * C_ + j0];          // B[j][i] = M[i=r][j]
        bv.y = Mh[r * C_ + j0 + 1];
        d = __builtin_amdgcn_wmma_f32_16x16x4_f32(
                false, av, false, bv, (short)0, d, false, false);
    }

    // Row norms: tree-reduce 8 D registers across the 16 columns.
    float t[8];
#pragma unroll
    for (int v = 0; v < 8; ++v) t[v] = d[v] * d[v];
    const float s = row_reduce16(t, r);

    const int   row = (lane & 7) | (hh << 3);   // row this lane now holds
    const float w   = ((vm >> row) & 1u) ? __expf(-s) : 0.f;

    // deg = sum of all 16 row weights (octet fold + cross-half fold)
    float dp = w;
    dp += SWZ(dp, 1);
    dp += SWZ(dp, 2);
    dp += SWZ(dp, 4);
    dp += SWZ(dp, 16);

    if ((lane & 15) < 8)
        w_out[(((size_t)bt * N_ + n) * 16 + row) * H_ + h] = w;
    if (lane == 0)
        deg[((size_t)bt * N_ + n) * H_ + h] = dp;
}

// ---------------------------------------------------------------------------
// Kernel 2: u_w = w * inv(sqrt(deg_i * deg_j)), deg_j gathered (0 if invalid)
// ---------------------------------------------------------------------------
__global__ void k2_uw(float* __restrict__ uw,
                      const float* __restrict__ deg,
                      const int*   __restrict__ nn)
{
    const size_t tid = (size_t)blockIdx.x * blockDim.x + threadIdx.x;
    const size_t total = (size_t)B_ * T_ * N_ * 16 * H_;
    if (tid >= total) return;
    const int h  = (int)(tid % H_);
    const int k  = (int)((tid / H_) % 16);
    const int n  = (int)((tid / ((size_t)H_ * 16)) % N_);
    const int bt = (int)( tid / ((size_t)H_ * 16 * N_));

    const float w  = uw[tid];
    const int   nb = nn[n * K1_ + 1 + k];
    const float di = deg[((size_t)bt * N_ + n) * H_ + h];
    const float dj = (nb >= 0) ? deg[((size_t)bt * N_ + nb) * H_ + h] : 0.f;
    const float dm = sqrtf(di * dj);
    uw[tid] = (dm > 0.f) ? (w / dm) : 0.f;
}

// ---------------------------------------------------------------------------
// Kernel 3: Qj[b,t,n,h,i] = sum_j multiQ2[h,i,j] * feat[b,t+1,n,h,j]
// ---------------------------------------------------------------------------
__global__ void k3_qj(const float* __restrict__ feat,
                      const float* __restrict__ q2,
                      float*       __restrict__ qj)
{
    const size_t tid = (size_t)blockIdx.x * blockDim.x + threadIdx.x;
    const size_t total = (size_t)B_ * T7_ * N_ * H_;
    if (tid >= total) return;
    const int h   = (int)(tid % H_);
    const int n   = (int)((tid / H_) % N_);
    const int bt7 = (int)( tid / ((size_t)H_ * N_));
    const int t   = bt7 % T7_;
    const int b   = bt7 / T7_;

    const float* f = feat + ((((size_t)(b * T_ + t + 1) * N_ + n) * H_ + h) * C_);
    const float* Q = q2 + h * NO_ * C_;
    float* o = qj + tid * NO_;
#pragma unroll
    for (int i = 0; i < NO_; ++i) {
        float s = 0.f;
#pragma unroll
        for (int j = 0; j < C_; ++j) s += Q[i * C_ + j] * f[j];
        o[i] = s;
    }
}

// ---------------------------------------------------------------------------
// Kernel 4: per (b,t,h) process tiles of 16 rows over e = n*17+k:
//   Qi = fi_gather @ padded(multiQ1[h])^T via 4 chained WMMAs,
//   wd = exp(-Qi . Qj[n]) * valid   -> d_w region
// ---------------------------------------------------------------------------
__global__ void k4_wd(const float* __restrict__ feat,
                      const int*   __restrict__ nn,
                      const float* __restrict__ q1,
                      const float* __restrict__ qj,
                      float*       __restrict__ dw)
{
    const int wid  = (blockIdx.x * blockDim.x + threadIdx.x) >> 5;  // grid exact
    const int lane = threadIdx.x & 31;

    const int tile = wid % TILES2_;
    const int r1   = wid / TILES2_;
    const int h    = r1 % H_;
    const int bt7  = r1 / H_;            // 0..13
    const int t    = bt7 % T7_;
    const int bq   = bt7 / T7_;
    const int btf  = bq * T_ + t;        // fi uses time t (pad[:, :-1])

    const int r  = lane & 15;
    const int hh = lane >> 4;

    const int  e     = tile * 16 + r;
    const bool rowOK = (e < E2_);
    const int  ec    = rowOK ? e : 0;
    const int  n     = ec / K1_;
    const int  k     = ec - n * K1_;
    const int  nbrv  = nn[n * K1_ + k];     // k==0 row is self (always >= 0)
    const bool pred  = rowOK && (nbrv >= 0);
    const unsigned vm = (unsigned)__ballot(pred);
    const int  jn    = pred ? nbrv : 0;
    const float sA   = pred ? 1.f : 0.f;    // invalid/pad rows -> zero A row

    const float* fi = feat + ((((size_t)btf * N_ + jn) * H_ + h) * C_);
    const float* Qh = q1 + h * NO_ * C_;
    const float sB  = (r < NO_) ? 1.f : 0.f;  // zero-pad output cols 8..15

    v8f d = {0.f,0.f,0.f,0.f,0.f,0.f,0.f,0.f};
#pragma unroll
    for (int c = 0; c < 4; ++c) {
        const int j0 = 4 * c + 2 * hh;
        v2f av, bv;
        av.x = sA * fi[j0];
        av.y = sA * fi[j0 + 1];
        bv.x = sB * Qh[(r & (NO_ - 1)) * C_ + j0];      // B[j][i] = Q1[i=r][j]
        bv.y = sB * Qh[(r & (NO_ - 1)) * C_ + j0 + 1];
        d = __builtin_amdgcn_wmma_f32_16x16x4_f32(
                false, av, false, bv, (short)0, d, false, false);
    }

    // t[v] = Qi[row_v][col] * Qj[n(row_v)][col], then tree-reduce over columns.
    float tv[8];
#pragma unroll
    for (int v = 0; v < 8; ++v) {
        const int ev = tile * 16 + v + (hh << 3);       // row in this lane half
        float qv = 0.f;
        if (ev < E2_ && r < NO_) {
            const int nv = ev / K1_;
            qv = qj[(((size_t)bt7 * N_ + nv) * H_ + h) * NO_ + r];
        }
        tv[v] = d[v] * qv;
    }
    const float s = row_reduce16(tv, r);

    const int   row = (lane & 7) | (hh << 3);
    const float wv  = ((vm >> row) & 1u) ? __expf(-s) : 0.f;

    if ((lane & 15) < 8) {
        const int ew = tile * 16 + row;
        if (ew < E2_)
            dw[((size_t)bt7 * E2_ + ew) * H_ + h] = wv;
    }
}

// ---------------------------------------------------------------------------
// Kernel 5: in-degree normalize d_w in place (sum over 17 neighbors)
// ---------------------------------------------------------------------------
__global__ void k5_dw(float* __restrict__ dw)
{
    const size_t tid = (size_t)blockIdx.x * blockDim.x + threadIdx.x;
    const size_t total = (size_t)B_ * T7_ * N_ * H_;
    if (tid >= total) return;
    const int h   = (int)(tid % H_);
    const int n   = (int)((tid / H_) % N_);
    const int bt7 = (int)( tid / ((size_t)H_ * N_));

    float* base = dw + ((size_t)bt7 * E2_ + (size_t)n * K1_) * H_ + h;
    float s = 0.f;
#pragma unroll
    for (int k = 0; k < K1_; ++k) s += base[k * H_];
    const float inv = (s > 0.f) ? (1.f / s) : 0.f;
#pragma unroll
    for (int k = 0; k < K1_; ++k) base[k * H_] *= inv;
}

// ---------------------------------------------------------------------------
extern "C" void kernel_launch(void* const* d_in, const int* in_sizes, int n_in,
                              void* d_out, int out_size, void* d_ws, size_t ws_size,
                              hipStream_t stream)
{
    const float* feat = (const float*)d_in[0];   // (B,T,N,H,C)
    const int*   nn   = (const int*)  d_in[1];   // (N,K1)
    const float* q1   = (const float*)d_in[2];   // (H,8,C)
    const float* q2   = (const float*)d_in[3];   // (H,8,C)
    const float* mm   = (const float*)d_in[4];   // (H,C,C)

    float* uw = (float*)d_out;                           // (B,T,N,16,H) = 5,120,000
    float* dw = uw + (size_t)B_ * T_ * N_ * 16 * H_;     // (B,7,N,17,H) = 4,760,000

    float* deg = (float*)d_ws;                           // B*T*N*H    = 320,000 f32
    float* qjw = deg + (size_t)B_ * T_ * N_ * H_;        // B*7*N*H*8  = 2,240,000 f32

    const int BLK = 256;

    // k1: one wave per (bt,h,n): 320,000 waves -> 40,000 blocks (exact, EXEC full)
    {
        const int waves = B_ * T_ * H_ * N_;
        k1_w_deg<<<waves * 32 / BLK, BLK, 0, stream>>>(feat, nn, mm, uw, deg);
    }
    // k2: elementwise normalize u_w
    {
        const size_t total = (size_t)B_ * T_ * N_ * 16 * H_;
        k2_uw<<<(unsigned)((total + BLK - 1) / BLK), BLK, 0, stream>>>(uw, deg, nn);
    }
    // k3: Qj precompute
    {
        const size_t total = (size_t)B_ * T7_ * N_ * H_;
        k3_qj<<<(unsigned)((total + BLK - 1) / BLK), BLK, 0, stream>>>(feat, q2, qjw);
    }
    // k4: one wave per 16-row tile: 14*4*5313 = 297,528 waves -> 37,191 blocks (exact)
    {
        const long long waves = (long long)B_ * T7_ * H_ * TILES2_;
        k4_wd<<<(unsigned)(waves * 32 / BLK), BLK, 0, stream>>>(feat, nn, q1, qjw, dw);
    }
    // k5: in-degree normalize d_w
    {
        const size_t total = (size_t)B_ * T7_ * N_ * H_;
        k5_dw<<<(unsigned)((total + BLK - 1) / BLK), BLK, 0, stream>>>(dw);
    }
}